// SparseAttention_22256520528655
// MI455X (gfx1250) — compile-verified
//
#include <hip/hip_runtime.h>

// Banded attention (|i-j| <= 128) for B=1, H=16, S=4096, D=64, fp32.
// d_out = [output 16*4096*64 floats] ++ [attn_weights 16*4096*4096 floats].
//
// MI455X reasoning: the 1 GiB dense attn_weights store dominates (≈46us at
// 23.3 TB/s). Q/K/V total 48 MB -> fully L2-resident (192 MB), so score-tile
// recompute across the 3 softmax passes costs only L2 hits. Matrix math uses
// exact-fp32 V_WMMA_F32_16X16X4_F32 (compute is ~4 GFLOP, negligible).

#define S_LEN  4096
#define HEADS  16
#define DDIM   64
#define WIN    128
#define NTILE  17          // (16 + 2*128)/16 key tiles per 16-row query block
#define LDSP   17          // padded LDS row stride (bank-conflict-free)

typedef float v2f __attribute__((ext_vector_type(2)));
typedef float v4f __attribute__((ext_vector_type(4)));
typedef float v8f __attribute__((ext_vector_type(8)));

__device__ __forceinline__ v8f wmma_f32(v2f a, v2f b, v8f c) {
  // D = A(16x4,f32) * B(4x16,f32) + C(16x16,f32)
  return __builtin_amdgcn_wmma_f32_16x16x4_f32(
      /*neg_a=*/false, a, /*neg_b=*/false, b,
      /*c_mod=*/(short)0, c, /*reuse_a=*/false, /*reuse_b=*/false);
}

// Compute one masked 16x16 score tile: S = (Q*scale) * K^T, band-masked.
// qa[c] is the A-operand chunk for K-dim [4c..4c+3] (pre-scaled).
__device__ __forceinline__ v8f score_tile(const v2f* qa, const float* __restrict__ Kh,
                                          int tbase, int qbase, int lane16, int hi) {
  v8f s = {0.f, 0.f, 0.f, 0.f, 0.f, 0.f, 0.f, 0.f};
  // B operand: lane L, vgpr r -> K[tbase + L%16][4c + 2*hi + r]  (contig float2)
  const float* kp = Kh + (size_t)(tbase + lane16) * DDIM + 2 * hi;
#pragma unroll
  for (int c = 0; c < 16; ++c) {
    v2f b = *(const v2f*)(kp + 4 * c);
    s = wmma_f32(qa[c], b, s);
  }
  // band mask: C layout -> vgpr r holds row (r + 8*hi), col = lane16
  int j = tbase + lane16;
#pragma unroll
  for (int r = 0; r < 8; ++r) {
    int i = qbase + r + 8 * hi;
    int d = i - j;
    if (d > WIN || d < -WIN) s[r] = -1e30f;
  }
  return s;
}

__global__ __launch_bounds__(128) void band_attn_wmma_kernel(
    const float* __restrict__ q, const float* __restrict__ k,
    const float* __restrict__ v, float* __restrict__ out) {
  __shared__ float lds[4 * 16 * LDSP];

  const int lane   = threadIdx.x & 31;
  const int wib    = threadIdx.x >> 5;     // wave in block (0..3)
  const int lane16 = lane & 15;
  const int hi     = lane >> 4;            // 0: lanes 0-15, 1: lanes 16-31
  const int waveId = blockIdx.x * 4 + wib; // 4096 waves total
  const int h      = waveId >> 8;          // head 0..15
  const int qbase  = (waveId & 255) * 16;  // query-row block base

  const float* Qh = q + (size_t)h * S_LEN * DDIM;
  const float* Kh = k + (size_t)h * S_LEN * DDIM;
  const float* Vh = v + (size_t)h * S_LEN * DDIM;
  float* Oh = out + (size_t)h * S_LEN * DDIM;
  float* Wh = out + (size_t)HEADS * S_LEN * DDIM + (size_t)h * S_LEN * S_LEN;
  float* myLds = lds + wib * (16 * LDSP);

  const float scale = 0.125f; // 1/sqrt(64)

  // ---- Load Q block in WMMA A-layout, pre-scaled ----------------------------
  // lane L, vgpr r, chunk c -> Q[qbase + L%16][4c + 2*hi + r]
  v2f qa[16];
  {
    const float* qp = Qh + (size_t)(qbase + lane16) * DDIM + 2 * hi;
#pragma unroll
    for (int c = 0; c < 16; ++c) {
      v2f t = *(const v2f*)(qp + 4 * c);
      t.x *= scale; t.y *= scale;
      qa[c] = t;
    }
  }

  // ---- Pass 1: row max ------------------------------------------------------
  float ml[8];
#pragma unroll
  for (int r = 0; r < 8; ++r) ml[r] = -1e30f;
  for (int t = 0; t < NTILE; ++t) {
    int tbase = qbase - WIN + t * 16;
    if (tbase < 0 || tbase >= S_LEN) continue;   // uniform branch: EXEC stays full
    v8f s = score_tile(qa, Kh, tbase, qbase, lane16, hi);
#pragma unroll
    for (int r = 0; r < 8; ++r) ml[r] = fmaxf(ml[r], s[r]);
  }
#pragma unroll
  for (int r = 0; r < 8; ++r) {
    float m0 = ml[r];
#pragma unroll
    for (int msk = 1; msk <= 8; msk <<= 1)       // reduce within 16-lane halves
      m0 = fmaxf(m0, __shfl_xor(m0, msk, 32));
    ml[r] = m0;
  }

  // ---- Pass 2: row sum of exp ----------------------------------------------
  float sl[8];
#pragma unroll
  for (int r = 0; r < 8; ++r) sl[r] = 0.f;
  for (int t = 0; t < NTILE; ++t) {
    int tbase = qbase - WIN + t * 16;
    if (tbase < 0 || tbase >= S_LEN) continue;
    v8f s = score_tile(qa, Kh, tbase, qbase, lane16, hi);
#pragma unroll
    for (int r = 0; r < 8; ++r) sl[r] += __expf(s[r] - ml[r]);
  }
#pragma unroll
  for (int r = 0; r < 8; ++r) {
    float s0 = sl[r];
#pragma unroll
    for (int msk = 1; msk <= 8; msk <<= 1)
      s0 += __shfl_xor(s0, msk, 32);
    sl[r] = 1.0f / s0;                           // store reciprocal
  }

  // ---- Pass 3: normalized weights out + P*V accumulation --------------------
  v8f o[4];
#pragma unroll
  for (int n = 0; n < 4; ++n) o[n] = (v8f){0.f, 0.f, 0.f, 0.f, 0.f, 0.f, 0.f, 0.f};

  for (int t = 0; t < NTILE; ++t) {
    int tbase = qbase - WIN + t * 16;
    if (tbase < 0 || tbase >= S_LEN) continue;
    v8f s = score_tile(qa, Kh, tbase, qbase, lane16, hi);
#pragma unroll
    for (int r = 0; r < 8; ++r)
      s[r] = __expf(s[r] - ml[r]) * sl[r];       // normalized probability

    // write band weights (streaming, never re-read)
#pragma unroll
    for (int r = 0; r < 8; ++r) {
      size_t idx = (size_t)(qbase + r + 8 * hi) * S_LEN + (size_t)(tbase + lane16);
      __builtin_nontemporal_store(s[r], &Wh[idx]);
    }

    // C-layout -> A-layout via padded LDS tile (per-wave private region)
#pragma unroll
    for (int r = 0; r < 8; ++r)
      myLds[(r + 8 * hi) * LDSP + lane16] = s[r];
    __asm__ volatile("s_wait_dscnt 0" ::: "memory");  // same-wave store->load

    // O += P(16x16) * V(16x64); A chunk: lane L -> P[L%16][4c + 2*hi + r]
#pragma unroll
    for (int c = 0; c < 4; ++c) {
      v2f a;
      a.x = myLds[lane16 * LDSP + 4 * c + 2 * hi];
      a.y = myLds[lane16 * LDSP + 4 * c + 2 * hi + 1];
#pragma unroll
      for (int n = 0; n < 4; ++n) {
        // B chunk: lane L, vgpr r -> V[tbase + 4c + 2*hi + r][16n + L%16]
        const float* vp = Vh + (size_t)(tbase + 4 * c + 2 * hi) * DDIM + n * 16 + lane16;
        v2f b; b.x = vp[0]; b.y = vp[DDIM];
        o[n] = wmma_f32(a, b, o[n]);
      }
    }
    __asm__ volatile("" ::: "memory");           // keep LDS reads before next writes
  }

  // ---- Store output block (C layout) ---------------------------------------
#pragma unroll
  for (int n = 0; n < 4; ++n)
#pragma unroll
    for (int r = 0; r < 8; ++r) {
      size_t idx = (size_t)(qbase + r + 8 * hi) * DDIM + (size_t)(n * 16 + lane16);
      __builtin_nontemporal_store(o[n][r], &Oh[idx]);
    }
}

// Bulk zero of the dense attn_weights region (band is overwritten afterwards).
__global__ __launch_bounds__(256) void zero_weights_kernel(v4f* __restrict__ w, long n4) {
  long i = (long)blockIdx.x * blockDim.x + threadIdx.x;
  long stride = (long)gridDim.x * blockDim.x;
  v4f z = {0.f, 0.f, 0.f, 0.f};
  for (; i < n4; i += stride)
    __builtin_nontemporal_store(z, &w[i]);
}

extern "C" void kernel_launch(void* const* d_in, const int* in_sizes, int n_in,
                              void* d_out, int out_size, void* d_ws, size_t ws_size,
                              hipStream_t stream) {
  (void)in_sizes; (void)n_in; (void)out_size; (void)d_ws; (void)ws_size;
  const float* q = (const float*)d_in[0];
  const float* k = (const float*)d_in[1];
  const float* v = (const float*)d_in[2];
  float* out = (float*)d_out;

  float* wpart = out + (size_t)HEADS * S_LEN * DDIM;
  long n4 = ((long)HEADS * S_LEN * S_LEN) / 4;   // 67,108,864 float4 = 1 GiB

  zero_weights_kernel<<<dim3(2048), dim3(256), 0, stream>>>((v4f*)wpart, n4);
  // 4096 waves = 16 heads x 256 query blocks; 4 waves per 128-thread block
  band_attn_wmma_kernel<<<dim3(1024), dim3(128), 0, stream>>>(q, k, v, out);
}